// GCNNet_10393820857081
// MI455X (gfx1250) — compile-verified
//
#include <hip/hip_runtime.h>
#include <hip/hip_bf16.h>

#define N_NODES  50000
#define N_EDGES  1600000
#define N_GRAPHS 64

typedef __attribute__((ext_vector_type(2))) float v2f;
typedef __attribute__((ext_vector_type(8))) float v8f;

// ---------------- degree / symmetric normalization (computed once, reused) ----
__global__ void k_init_deg(float* deg, int n) {
  int i = blockIdx.x * blockDim.x + threadIdx.x;
  if (i < n) deg[i] = 1.0f;                 // self-loop weight = 1
}

__global__ void k_accum_deg(const int* __restrict__ col, const float* __restrict__ ew,
                            float* deg, int nE) {
  int e = blockIdx.x * blockDim.x + threadIdx.x;
  if (e < nE) atomicAdd(&deg[col[e]], ew[e]);
}

__global__ void k_rsqrt_deg(float* deg, int n) {
  int i = blockIdx.x * blockDim.x + threadIdx.x;
  if (i < n) {
    float d = deg[i];
    deg[i] = d > 0.f ? rsqrtf(d) : 0.f;
  }
}

// ---------------- fp32 WMMA GEMM: H[M,FOUT] = X[M,K] @ W[K,FOUT] -------------
// One wave32 per 16-row stripe; all FOUT/16 column tiles accumulated in regs so
// the A fragment is loaded once per k-step. Exact fp32 (V_WMMA_F32_16X16X4_F32).
template<int K, int FOUT>
__global__ __launch_bounds__(32)
void k_gemm_wmma_f32(const float* __restrict__ X, const float* __restrict__ W,
                     float* __restrict__ H) {
  constexpr int NT = FOUT / 16;
  const int m0    = blockIdx.x * 16;
  const int lane  = threadIdx.x & 31;
  const int lrow  = lane & 15;             // M (for A) / N (for B,C) within tile
  const int khalf = (lane >> 4) * 2;       // lanes 0-15 hold K=k0,k0+1; 16-31 hold k0+2,k0+3

  v8f acc[NT];
#pragma unroll
  for (int t = 0; t < NT; ++t) acc[t] = {};

  const float* xr = X + (size_t)(m0 + lrow) * K + khalf;
#pragma unroll 4
  for (int k0 = 0; k0 < K; k0 += 4) {
    v2f a;
    a.x = xr[k0];
    a.y = xr[k0 + 1];
    const float* wr = W + (size_t)(k0 + khalf) * FOUT + lrow;
#pragma unroll
    for (int t = 0; t < NT; ++t) {
      v2f b;
      b.x = wr[t * 16];                    // row k0+khalf
      b.y = wr[t * 16 + FOUT];             // row k0+khalf+1
      acc[t] = __builtin_amdgcn_wmma_f32_16x16x4_f32(
          false, a, false, b, (short)0, acc[t], false, false);
    }
  }

  const int rbase = m0 + ((lane >> 4) * 8);   // C/D: VGPR j = rows j (lo lanes), j+8 (hi lanes)
#pragma unroll
  for (int t = 0; t < NT; ++t) {
#pragma unroll
    for (int j = 0; j < 8; ++j) {
      H[(size_t)(rbase + j) * FOUT + t * 16 + lrow] = acc[t][j];
    }
  }
}

// ---------------- self-loop term: out[i,:] = dinv[i]^2 * h[i,:] ---------------
template<int F>
__global__ void k_self_init(const float* __restrict__ h, const float* __restrict__ dinv,
                            float* __restrict__ out, int n) {
  int i = blockIdx.x * blockDim.x + threadIdx.x;
  if (i < n * F) {
    float d = dinv[i / F];
    out[i] = d * d * h[i];
  }
}

// ---------------- edge scatter-add: out[c,:] += dinv[r]*ew*dinv[c] * h[r,:] ---
// One wave32 per edge; lane owns F/32 contiguous floats (coalesced 128B/256B per
// half-wave). Grid-stride with prefetch of the next edge's source row (L2-resident).
template<int F>
__global__ void k_edge_aggregate(const float* __restrict__ h,
                                 const int* __restrict__ row,
                                 const int* __restrict__ col,
                                 const float* __restrict__ ew,
                                 const float* __restrict__ dinv,
                                 float* __restrict__ out, int nE) {
  constexpr int VEC = F / 32;
  const int lane   = threadIdx.x & 31;
  const int wpb    = blockDim.x >> 5;
  const int wave   = blockIdx.x * wpb + (threadIdx.x >> 5);
  const int stride = gridDim.x * wpb;
  for (int e = wave; e < nE; e += stride) {
    int en = e + stride;
    if (en < nE) {
      __builtin_prefetch(h + (size_t)row[en] * F + lane * VEC, 0, 1);
    }
    const int r = row[e], c = col[e];
    const float norm = dinv[r] * ew[e] * dinv[c];
    const float* src = h   + (size_t)r * F + lane * VEC;
    float*       dst = out + (size_t)c * F + lane * VEC;
#pragma unroll
    for (int v = 0; v < VEC; ++v) atomicAdd(dst + v, norm * src[v]);
  }
}

// ---------------- bias + relu (also the sync point after atomics) -------------
template<int F>
__global__ void k_bias_relu(float* __restrict__ out, const float* __restrict__ b, int n) {
  int i = blockIdx.x * blockDim.x + threadIdx.x;
  if (i < n * F) {
    float v = out[i] + b[i & (F - 1)];
    out[i] = fmaxf(v, 0.f);
  }
}

// ---------------- global mean pool ------------------------------------------
__global__ void k_zero(float* p, int n) {
  int i = blockIdx.x * blockDim.x + threadIdx.x;
  if (i < n) p[i] = 0.f;
}

__global__ void k_pool(const float* __restrict__ x1, const int* __restrict__ batch,
                       float* __restrict__ sums, float* __restrict__ cnt, int n) {
  const int lane = threadIdx.x & 31;
  const int node = blockIdx.x * (blockDim.x >> 5) + (threadIdx.x >> 5);
  if (node >= n) return;
  const int b = batch[node];
  const float* src = x1 + (size_t)node * 64 + lane * 2;
  atomicAdd(&sums[b * 64 + lane * 2 + 0], src[0]);
  atomicAdd(&sums[b * 64 + lane * 2 + 1], src[1]);
  if (lane == 0) atomicAdd(&cnt[b], 1.0f);
}

// ---------------- MLP head + log_softmax (single block, LDS-resident) --------
__global__ __launch_bounds__(256)
void k_head(const float* __restrict__ sums, const float* __restrict__ cnt,
            const float* __restrict__ fc1W, const float* __restrict__ fc1b,
            const float* __restrict__ fc2W, const float* __restrict__ fc2b,
            float* __restrict__ out) {
  __shared__ float x2[N_GRAPHS * 64];
  __shared__ float hbuf[N_GRAPHS * 128];
  __shared__ float logits[N_GRAPHS * 2];
  const int t = threadIdx.x;

  for (int i = t; i < N_GRAPHS * 64; i += 256) {
    int g = i >> 6;
    x2[i] = sums[i] / fmaxf(cnt[g], 1.0f);
  }
  __syncthreads();

  for (int i = t; i < N_GRAPHS * 128; i += 256) {
    int g = i >> 7, k = i & 127;
    float acc = fc1b[k];
#pragma unroll 8
    for (int j = 0; j < 64; ++j) acc += x2[g * 64 + j] * fc1W[j * 128 + k];
    hbuf[i] = fmaxf(acc, 0.f);
  }
  __syncthreads();

  if (t < N_GRAPHS * 2) {
    int g = t >> 1, c = t & 1;
    float acc = fc2b[c];
#pragma unroll 8
    for (int k = 0; k < 128; ++k) acc += hbuf[g * 128 + k] * fc2W[k * 2 + c];
    logits[t] = acc;
  }
  __syncthreads();

  if (t < N_GRAPHS * 2) {
    int g = t >> 1;
    float l0 = logits[g * 2], l1 = logits[g * 2 + 1];
    float m  = fmaxf(l0, l1);
    float lse = m + logf(expf(l0 - m) + expf(l1 - m));
    out[t] = logits[t] - lse;
  }
}

// ---------------- driver -----------------------------------------------------
extern "C" void kernel_launch(void* const* d_in, const int* in_sizes, int n_in,
                              void* d_out, int out_size, void* d_ws, size_t ws_size,
                              hipStream_t stream) {
  (void)in_sizes; (void)n_in; (void)out_size; (void)ws_size;
  const float* x    = (const float*)d_in[0];
  const int*   ei   = (const int*)d_in[1];
  const float* ew   = (const float*)d_in[2];
  const int*   batch= (const int*)d_in[3];
  const float* W1   = (const float*)d_in[4];
  const float* b1   = (const float*)d_in[5];
  const float* W2   = (const float*)d_in[6];
  const float* b2   = (const float*)d_in[7];
  const float* fc1W = (const float*)d_in[8];
  const float* fc1b = (const float*)d_in[9];
  const float* fc2W = (const float*)d_in[10];
  const float* fc2b = (const float*)d_in[11];
  const int* row = ei;
  const int* col = ei + N_EDGES;

  // workspace layout (floats, 256B-aligned chunks), ~77 MB total
  float* ws   = (float*)d_ws;
  size_t o    = 0;
  float* dinv = ws + o;  o += 50048;                    // N rounded to 64
  float* h1   = ws + o;  o += (size_t)N_NODES * 128;
  float* out1 = ws + o;  o += (size_t)N_NODES * 128;
  float* h2   = ws + o;  o += (size_t)N_NODES * 64;
  float* out2 = ws + o;  o += (size_t)N_NODES * 64;
  float* sums = ws + o;  o += N_GRAPHS * 64;
  float* cnt  = ws + o;  o += N_GRAPHS;

  // normalization (shared by both conv layers)
  k_init_deg  <<<(N_NODES + 255) / 256, 256, 0, stream>>>(dinv, N_NODES);
  k_accum_deg <<<(N_EDGES + 255) / 256, 256, 0, stream>>>(col, ew, dinv, N_EDGES);
  k_rsqrt_deg <<<(N_NODES + 255) / 256, 256, 0, stream>>>(dinv, N_NODES);

  // conv1: h1 = x @ W1 ; out1 = relu(Aggregate(h1) + b1)
  k_gemm_wmma_f32<128,128><<<N_NODES / 16, 32, 0, stream>>>(x, W1, h1);
  k_self_init<128><<<(N_NODES * 128 + 255) / 256, 256, 0, stream>>>(h1, dinv, out1, N_NODES);
  k_edge_aggregate<128><<<20000, 256, 0, stream>>>(h1, row, col, ew, dinv, out1, N_EDGES);
  k_bias_relu<128><<<(N_NODES * 128 + 255) / 256, 256, 0, stream>>>(out1, b1, N_NODES);

  // conv2: h2 = out1 @ W2 ; out2 = relu(Aggregate(h2) + b2)
  k_gemm_wmma_f32<128,64><<<N_NODES / 16, 32, 0, stream>>>(out1, W2, h2);
  k_self_init<64><<<(N_NODES * 64 + 255) / 256, 256, 0, stream>>>(h2, dinv, out2, N_NODES);
  k_edge_aggregate<64><<<20000, 256, 0, stream>>>(h2, row, col, ew, dinv, out2, N_EDGES);
  k_bias_relu<64><<<(N_NODES * 64 + 255) / 256, 256, 0, stream>>>(out2, b2, N_NODES);

  // mean pool + head
  k_zero<<<(N_GRAPHS * 64 + N_GRAPHS + 255) / 256, 256, 0, stream>>>(sums, N_GRAPHS * 64 + N_GRAPHS);
  k_pool<<<(N_NODES + 7) / 8, 256, 0, stream>>>(out2, batch, sums, cnt, N_NODES);
  k_head<<<1, 256, 0, stream>>>(sums, cnt, fc1W, fc1b, fc2W, fc2b, (float*)d_out);
}